// HistogramEqualization_10453950398758
// MI455X (gfx1250) — compile-verified
//
#include <hip/hip_runtime.h>

#define NBINS 256

typedef __attribute__((ext_vector_type(2))) float v2f;
typedef __attribute__((ext_vector_type(8))) float v8f;

// ---------------------------------------------------------------------------
// Kernel 0: zero the global histogram (256 x u32 in workspace)
// ---------------------------------------------------------------------------
__global__ __launch_bounds__(256) void he_zero_hist(unsigned int* __restrict__ ghist) {
    ghist[threadIdx.x] = 0u;
}

// ---------------------------------------------------------------------------
// Kernel 1: histogram. Per-wave LDS sub-histograms (wave32 -> 8 waves/block),
// ds_add_u32 atomics, one global_atomic_add_u32 merge per block per bin.
// Processes float4 (global_load_b128) for bandwidth.
// ---------------------------------------------------------------------------
__global__ __launch_bounds__(256) void he_hist(const float* __restrict__ x,
                                               unsigned int* __restrict__ ghist,
                                               int n4, int ntail) {
    __shared__ unsigned int lh[8][NBINS];   // 8 KB: one copy per wave32
    // zero all copies
    for (int b = threadIdx.x; b < 8 * NBINS; b += blockDim.x)
        ((unsigned int*)lh)[b] = 0u;
    __syncthreads();

    unsigned int* h = lh[threadIdx.x >> 5];

    const float4* x4 = (const float4*)x;
    const int stride = gridDim.x * blockDim.x;
    for (int i = blockIdx.x * blockDim.x + threadIdx.x; i < n4; i += stride) {
        __builtin_prefetch(&x4[i + stride], 0, 0);   // global_prefetch_b8
        float4 v = x4[i];
        int b0 = min(max((int)(v.x * 256.0f), 0), NBINS - 1);
        int b1 = min(max((int)(v.y * 256.0f), 0), NBINS - 1);
        int b2 = min(max((int)(v.z * 256.0f), 0), NBINS - 1);
        int b3 = min(max((int)(v.w * 256.0f), 0), NBINS - 1);
        atomicAdd(&h[b0], 1u);   // ds_add_u32
        atomicAdd(&h[b1], 1u);
        atomicAdd(&h[b2], 1u);
        atomicAdd(&h[b3], 1u);
    }
    // scalar tail (n not divisible by 4) handled by block 0
    if (blockIdx.x == 0 && (int)threadIdx.x < ntail) {
        float v = x[n4 * 4 + threadIdx.x];
        int b = min(max((int)(v * 256.0f), 0), NBINS - 1);
        atomicAdd(&h[b], 1u);
    }
    __syncthreads();

    // reduce the 8 wave-private copies, merge to global
    for (int b = threadIdx.x; b < NBINS; b += blockDim.x) {
        unsigned int s = 0u;
        #pragma unroll
        for (int w = 0; w < 8; ++w) s += lh[w][b];
        atomicAdd(&ghist[b], s);   // global_atomic_add_u32
    }
}

// ---------------------------------------------------------------------------
// Kernel 2: cumsum of 256 bins + normalization via WMMA.
// One wave (32 threads, EXEC all ones as WMMA requires).
//   H = hist as 16x16 row-major (f32).
//   cum = H*U + L'*S   (U upper-tri incl. ones; L' strict lower-tri ones;
//                       S[k][n] = rowsum(H[k]))
// Both products accumulate into one v8f C via 8x V_WMMA_F32_16X16X4_F32.
// ---------------------------------------------------------------------------
__global__ __launch_bounds__(32) void he_cdf_wmma(const unsigned int* __restrict__ ghist,
                                                  float* __restrict__ cdf) {
    __shared__ float h[NBINS];
    __shared__ float rowsum[16];
    __shared__ float cum[NBINS];

    const int lane = threadIdx.x;     // 0..31
    const int half = lane >> 4;       // 0 | 1
    const int m    = lane & 15;       // row (A) / col (B,C,D)

    for (int b = lane; b < NBINS; b += 32) h[b] = (float)ghist[b];
    __syncthreads();

    if (lane < 16) {
        float s = 0.0f;
        #pragma unroll
        for (int c = 0; c < 16; ++c) s += h[lane * 16 + c];
        rowsum[lane] = s;
    }
    __syncthreads();

    v8f acc = {};

    // chain 1: R = H * U     A[M][K] = H[m][k], B[K][N] = (k <= m)
    #pragma unroll
    for (int q = 0; q < 4; ++q) {
        const int k0 = 4 * q + 0 + 2 * half;
        const int k1 = 4 * q + 1 + 2 * half;
        v2f a, bb;
        a.x  = h[m * 16 + k0];
        a.y  = h[m * 16 + k1];
        bb.x = (k0 <= m) ? 1.0f : 0.0f;
        bb.y = (k1 <= m) ? 1.0f : 0.0f;
        acc = __builtin_amdgcn_wmma_f32_16x16x4_f32(false, a, false, bb,
                                                    (short)0, acc, false, false);
    }
    // chain 2: T = L' * S    A[M][K] = (k < m), B[K][N] = rowsum[k]
    #pragma unroll
    for (int q = 0; q < 4; ++q) {
        const int k0 = 4 * q + 0 + 2 * half;
        const int k1 = 4 * q + 1 + 2 * half;
        v2f a, bb;
        a.x  = (k0 < m) ? 1.0f : 0.0f;
        a.y  = (k1 < m) ? 1.0f : 0.0f;
        bb.x = rowsum[k0];
        bb.y = rowsum[k1];
        acc = __builtin_amdgcn_wmma_f32_16x16x4_f32(false, a, false, bb,
                                                    (short)0, acc, false, false);
    }

    // scatter C/D layout (VGPR r, lane) -> bin = (r + 8*half)*16 + m
    #pragma unroll
    for (int r = 0; r < 8; ++r)
        cum[(r + 8 * half) * 16 + m] = acc[r];
    __syncthreads();

    const float c0   = cum[0];
    const float cmax = cum[NBINS - 1];
    const float inv  = 1.0f / (cmax - c0);
    for (int b = lane; b < NBINS; b += 32)
        cdf[b] = (cum[b] - c0) * inv;
}

// ---------------------------------------------------------------------------
// Kernel 3: per-element interp against the 256-entry CDF (staged in LDS).
// out = cdf[i] + (v - i) * (cdf[i+1] - cdf[i]),  i = clamp(floor(v), 0, 254)
// ---------------------------------------------------------------------------
__device__ __forceinline__ float he_interp(float v, const float* __restrict__ cdf) {
    int i = (int)floorf(v);
    i = min(max(i, 0), NBINS - 2);
    float f = v - (float)i;
    float c = cdf[i];                 // ds_load
    return c + f * (cdf[i + 1] - c);
}

__global__ __launch_bounds__(256) void he_map(const float* __restrict__ x,
                                              const float* __restrict__ cdf,
                                              float* __restrict__ out,
                                              int n4, int ntail) {
    __shared__ float lcdf[NBINS];
    for (int b = threadIdx.x; b < NBINS; b += blockDim.x) lcdf[b] = cdf[b];
    __syncthreads();

    const float4* x4 = (const float4*)x;
    float4* o4 = (float4*)out;
    const int stride = gridDim.x * blockDim.x;
    for (int i = blockIdx.x * blockDim.x + threadIdx.x; i < n4; i += stride) {
        __builtin_prefetch(&x4[i + stride], 0, 0);
        float4 v = x4[i];
        float4 r;
        r.x = he_interp(v.x, lcdf);
        r.y = he_interp(v.y, lcdf);
        r.z = he_interp(v.z, lcdf);
        r.w = he_interp(v.w, lcdf);
        o4[i] = r;                     // global_store_b128
    }
    if (blockIdx.x == 0 && (int)threadIdx.x < ntail) {
        int i = n4 * 4 + threadIdx.x;
        out[i] = he_interp(x[i], lcdf);
    }
}

// ---------------------------------------------------------------------------
extern "C" void kernel_launch(void* const* d_in, const int* in_sizes, int n_in,
                              void* d_out, int out_size, void* d_ws, size_t ws_size,
                              hipStream_t stream) {
    const float* x = (const float*)d_in[0];
    float* out = (float*)d_out;
    const int n  = in_sizes[0];           // 32*3*512*512 = 25,165,824
    const int n4 = n >> 2;
    const int ntail = n & 3;

    unsigned int* ghist = (unsigned int*)d_ws;                         // 256 u32
    float* cdf = (float*)((char*)d_ws + NBINS * sizeof(unsigned int)); // 256 f32

    // grid-stride sizing: ~16 float4s per thread, capped
    int blocks = (n4 + 256 * 16 - 1) / (256 * 16);
    if (blocks < 1) blocks = 1;
    if (blocks > 4096) blocks = 4096;

    he_zero_hist<<<1, 256, 0, stream>>>(ghist);
    he_hist<<<blocks, 256, 0, stream>>>(x, ghist, n4, ntail);
    he_cdf_wmma<<<1, 32, 0, stream>>>(ghist, cdf);
    he_map<<<blocks, 256, 0, stream>>>(x, cdf, out, n4, ntail);
}